// RWKV_RNN_5274219840223
// MI455X (gfx1250) — compile-verified
//
#include <hip/hip_runtime.h>

#define DEVFN __device__ __forceinline__

namespace {
constexpr int Cc  = 1024;
constexpr int Hh  = 16;
constexpr int Nn  = 64;
constexpr int Ll  = 24;
constexpr int DLr = 64;
constexpr int FFd = 4096;
constexpr int Vv  = 65536;

// d_out offsets (floats): [logits V][new_x1 L*C][new_S L*H*N*N][new_x2 L*C]
constexpr size_t O_LOGITS = 0;
constexpr size_t O_X1 = (size_t)Vv;
constexpr size_t O_S  = O_X1 + (size_t)Ll * Cc;
constexpr size_t O_X2 = O_S + (size_t)Ll * Hh * Nn * Nn;

// workspace offsets (floats)
constexpr size_t W_X = 0, W_XR = 1024, W_XK = 2048, W_XV = 3072, W_XW = 4096,
                 W_XA = 5120, W_XG = 6144, W_WD = 7168, W_A = 8192, W_VVS = 9216,
                 W_G = 10240, W_R = 11264, W_K = 12288, W_V = 13312, W_KK = 14336,
                 W_CV = 15360, W_KADJ = 16384, W_VCUR = 17408, W_VF = 18432,
                 W_OG = 19456, W_KFIN = 20480, W_KF = 21504, W_XH = 25600,
                 W_RKS = 26624, W_T1 = 26656;   // total ~27k floats (~108 KB)
} // namespace

typedef float v2f_t __attribute__((ext_vector_type(2)));
typedef float v8f_t __attribute__((ext_vector_type(8)));

DEVFN float sigm(float x) { return 1.0f / (1.0f + expf(-x)); }

// Block-wide (256 threads) mean / rstd over C=1024 (4 values per thread).
DEVFN void ln_stats_256(float s, float ss, float* red, float& mu, float& rs, float eps) {
  const int t = threadIdx.x;
  red[t] = s; __syncthreads();
  for (int o = 128; o > 0; o >>= 1) { if (t < o) red[t] += red[t + o]; __syncthreads(); }
  const float tot = red[0]; __syncthreads();
  red[t] = ss; __syncthreads();
  for (int o = 128; o > 0; o >>= 1) { if (t < o) red[t] += red[t + o]; __syncthreads(); }
  const float totq = red[0]; __syncthreads();
  mu = tot * (1.0f / Cc);
  const float var = totq * (1.0f / Cc) - mu * mu;
  rs = rsqrtf(var + eps);
}

// x = ln(emb[token], ln0_w, ln0_b)
__global__ void k_embed(const int* __restrict__ token, const float* __restrict__ emb,
                        const float* __restrict__ w, const float* __restrict__ b,
                        float* __restrict__ ws) {
  __shared__ float red[256];
  const int t = threadIdx.x;
  const float* row = emb + (size_t)token[0] * Cc;
  float v[4], s = 0.f, ss = 0.f;
#pragma unroll
  for (int q = 0; q < 4; q++) { v[q] = row[t * 4 + q]; s += v[q]; ss += v[q] * v[q]; }
  float mu, rs; ln_stats_256(s, ss, red, mu, rs, 1e-5f);
#pragma unroll
  for (int q = 0; q < 4; q++) { int c = t * 4 + q; ws[W_X + c] = (v[q] - mu) * rs * w[c] + b[c]; }
}

// ln1 + token-shift mixes; writes new_x1 and xr/xk/xv/xw/xa/xg
__global__ void k_prep(int l, const float* __restrict__ ln1_w, const float* __restrict__ ln1_b,
                       const float* __restrict__ s_x1,
                       const float* __restrict__ mx_r, const float* __restrict__ mx_w,
                       const float* __restrict__ mx_k, const float* __restrict__ mx_v,
                       const float* __restrict__ mx_a, const float* __restrict__ mx_g,
                       float* __restrict__ out_x1, float* __restrict__ ws) {
  __shared__ float red[256];
  const int t = threadIdx.x;
  float v[4], s = 0.f, ss = 0.f;
#pragma unroll
  for (int q = 0; q < 4; q++) { v[q] = ws[W_X + t * 4 + q]; s += v[q]; ss += v[q] * v[q]; }
  float mu, rs; ln_stats_256(s, ss, red, mu, rs, 1e-5f);
  const size_t lC = (size_t)l * Cc;
#pragma unroll
  for (int q = 0; q < 4; q++) {
    const int c = t * 4 + q;
    const float xx = (v[q] - mu) * rs * ln1_w[lC + c] + ln1_b[lC + c];
    out_x1[lC + c] = xx;
    const float d = s_x1[lC + c] - xx;
    ws[W_XR + c] = xx + d * mx_r[lC + c];
    ws[W_XK + c] = xx + d * mx_k[lC + c];
    ws[W_XV + c] = xx + d * mx_v[lC + c];
    ws[W_XW + c] = xx + d * mx_w[lC + c];
    ws[W_XA + c] = xx + d * mx_a[lC + c];
    ws[W_XG + c] = xx + d * mx_g[lC + c];
  }
}

// LoRA stage 1: t1 = act(xvec @ M1), M1 is [C, DL]; block = chain, 512 threads.
__global__ void k_lora1(int l, const float* __restrict__ w1, const float* __restrict__ a1,
                        const float* __restrict__ v1, const float* __restrict__ g1,
                        float* __restrict__ ws) {
  const int chain = blockIdx.x;
  const float* M1 = (chain == 0 ? w1 : chain == 1 ? a1 : chain == 2 ? v1 : g1) + (size_t)l * Cc * DLr;
  const float* x = ws + (chain == 0 ? W_XW : chain == 1 ? W_XA : chain == 2 ? W_XV : W_XG);
  const int t = threadIdx.x;       // 512
  const int dl = t & 63, cg = t >> 6;  // 8 c-groups x 64 dl
  float acc = 0.f;
  const int c0 = cg * 128;
  for (int c = c0; c < c0 + 128; c++) acc += x[c] * M1[(size_t)c * DLr + dl];
  __shared__ float partial[8][64];
  partial[cg][dl] = acc;
  __syncthreads();
  if (cg == 0) {
    float sum = 0.f;
#pragma unroll
    for (int q = 0; q < 8; q++) sum += partial[q][dl];
    if (chain == 0) sum = tanhf(sum);
    else if (chain == 3) sum = sigm(sum);
    ws[W_T1 + chain * 64 + dl] = sum;
  }
}

// LoRA stage 2 + nonlinearity: wdec / a / v-sigmoid / g vectors. 16 blocks x 256.
__global__ void k_lora2(int l, const float* __restrict__ w2, const float* __restrict__ a2,
                        const float* __restrict__ v2, const float* __restrict__ g2,
                        const float* __restrict__ w0, const float* __restrict__ a0,
                        const float* __restrict__ v0, float* __restrict__ ws) {
  const int idx = blockIdx.x * 256 + threadIdx.x;  // 0..4095
  const int chain = idx >> 10, c = idx & (Cc - 1);
  const float* M2 = (chain == 0 ? w2 : chain == 1 ? a2 : chain == 2 ? v2 : g2) + (size_t)l * DLr * Cc;
  float acc = 0.f;
#pragma unroll
  for (int d = 0; d < DLr; d++) acc += ws[W_T1 + chain * 64 + d] * M2[(size_t)d * Cc + c];
  const size_t lC = (size_t)l * Cc;
  if (chain == 0)      ws[W_WD + c] = expf(-0.606531f * sigm(w0[lC + c] + acc));
  else if (chain == 1) ws[W_A + c]  = sigm(a0[lC + c] + acc);
  else if (chain == 2) ws[W_VVS + c] = sigm(v0[lC + c] + acc);
  else                 ws[W_G + c]  = acc;  // sigmoid already applied in stage 1
}

// Fused r/k/v GEMVs: 3072 rows, one wave per row, float4 coalesced.
__global__ void k_gemv3(const float* __restrict__ Wr, const float* __restrict__ Wk,
                        const float* __restrict__ Wv, float* __restrict__ ws) {
  const int wid = (blockIdx.x * blockDim.x + threadIdx.x) >> 5;
  const int lane = threadIdx.x & 31;
  const int seg = wid >> 10, row = wid & (Cc - 1);
  const float* W = seg == 0 ? Wr : (seg == 1 ? Wk : Wv);
  const float* x = ws + (seg == 0 ? W_XR : (seg == 1 ? W_XK : W_XV));
  float* y = ws + (seg == 0 ? W_R : (seg == 1 ? W_K : W_V));
  const float4* W4 = (const float4*)W + (size_t)row * (Cc / 4);
  const float4* x4 = (const float4*)x;
  float acc = 0.f;
  for (int t2 = lane; t2 < Cc / 4; t2 += 32) {
    const float4 a = W4[t2], b = x4[t2];
    acc += a.x * b.x + a.y * b.y + a.z * b.z + a.w * b.w;
  }
#pragma unroll
  for (int m = 16; m > 0; m >>= 1) acc += __shfl_xor(acc, m, 32);
  if (lane == 0) y[row] = acc;
}

// Generic wave-per-row GEMV. mode 0: y=dot; 1: y=add+dot; 2: y=relu(dot)^2.
__global__ void k_gemv(const float* __restrict__ W, const float* __restrict__ x,
                       const float* __restrict__ add, float* __restrict__ y,
                       int K4, int mode) {
  const int wid = (blockIdx.x * blockDim.x + threadIdx.x) >> 5;
  const int lane = threadIdx.x & 31;
  const float4* W4 = (const float4*)W + (size_t)wid * K4;
  const float4* x4 = (const float4*)x;
  float acc = 0.f;
  for (int t2 = lane; t2 < K4; t2 += 32) {
    const float4 a = W4[t2], b = x4[t2];
    acc += a.x * b.x + a.y * b.y + a.z * b.z + a.w * b.w;
  }
#pragma unroll
  for (int m = 16; m > 0; m >>= 1) acc += __shfl_xor(acc, m, 32);
  if (lane == 0) {
    if (mode == 1) acc += add[wid];
    else if (mode == 2) { acc = fmaxf(acc, 0.f); acc = acc * acc; }
    y[wid] = acc;
  }
}

// kk normalization per head, k adjust, v_first mixing, r*k*r_k per-head sums.
__global__ void k_post1(int l, const float* __restrict__ k_k, const float* __restrict__ k_a,
                        const float* __restrict__ r_k, float* __restrict__ ws) {
  const int t = threadIdx.x;  // 256; head = t>>4 (16 lanes x 4 channels)
  const size_t lC = (size_t)l * Cc;
  const int base = t * 4;
  float kkraw[4]; float sq = 0.f;
#pragma unroll
  for (int q = 0; q < 4; q++) {
    const int c = base + q;
    const float kv = ws[W_K + c] * k_k[lC + c];
    kkraw[q] = kv; sq += kv * kv;
  }
#pragma unroll
  for (int m = 1; m < 16; m <<= 1) sq += __shfl_xor(sq, m, 32);  // reduce within head's 16 lanes
  const float inv = 1.0f / fmaxf(sqrtf(sq), 1e-12f);
  float rkp = 0.f;
#pragma unroll
  for (int q = 0; q < 4; q++) {
    const int c = base + q;
    const float kkn = kkraw[q] * inv;   ws[W_KK + c] = kkn;
    const float av = ws[W_A + c];       ws[W_CV + c] = kkn * av;
    const float kadj = ws[W_K + c] * (1.f + (av - 1.f) * k_a[lC + c]);
    ws[W_KADJ + c] = kadj;
    const float vraw = ws[W_V + c];
    float vc;
    if (l == 0) { vc = vraw; ws[W_VF + c] = vraw; }
    else        { vc = vraw + (ws[W_VF + c] - vraw) * ws[W_VVS + c]; }
    ws[W_VCUR + c] = vc;
    rkp += ws[W_R + c] * kadj * r_k[lC + c];
  }
#pragma unroll
  for (int m = 1; m < 16; m <<= 1) rkp += __shfl_xor(rkp, m, 32);
  if ((t & 15) == 0) ws[W_RKS + (t >> 4)] = rkp;
}

// State update via WMMA f32 16x16x4: S' = S*diag(w) + S@ab + v (x) k, then
// out = S'@r, per-head group norm, lnx affine, +rk_sum*v, *g  -> og vector.
// Grid: 16 blocks (one per head) x 128 threads (4 waves, 16 rows each).
__global__ void k_state(int l, const float* __restrict__ s_S,
                        const float* __restrict__ lnx_w, const float* __restrict__ lnx_b,
                        float* __restrict__ S_out, float* __restrict__ ws) {
  const int h = blockIdx.x;
  const int t = threadIdx.x;
  const int wave = t >> 5;
  const int lane = t & 31;
  const int half = lane >> 4;    // K-half for A/B fragments
  const int lr = lane & 15;      // row (A) / column (B,D) index within tile
  const int i0 = wave * 16;

  const float* Sh = s_S + ((size_t)l * Hh + h) * Nn * Nn;
  float*       So = S_out + ((size_t)l * Hh + h) * Nn * Nn;
  const float* kkh = ws + W_KK + h * Nn;
  const float* cvh = ws + W_CV + h * Nn;    // kk[j]*a[j]
  const float* wdh = ws + W_WD + h * Nn;
  const float* kah = ws + W_KADJ + h * Nn;
  const float* rh  = ws + W_R + h * Nn;
  const float* vch = ws + W_VCUR + h * Nn;

  float cv[4], wv[4], kv[4], rv[4];
#pragma unroll
  for (int tt = 0; tt < 4; tt++) {
    const int j = tt * 16 + lr;
    cv[tt] = cvh[j]; wv[tt] = wdh[j]; kv[tt] = kah[j]; rv[tt] = rh[j];
  }

  v8f_t acc[4];
#pragma unroll
  for (int tt = 0; tt < 4; tt++) {
#pragma unroll
    for (int p = 0; p < 8; p++) acc[tt][p] = 0.f;
  }

  const int arow = i0 + lr;
  const int kofs = half * 2;
  for (int ks = 0; ks < Nn; ks += 4) {
    // A fragment: S[arow][ks+kofs .. +1] (fp32 16x4 layout)
    const float2 av = *reinterpret_cast<const float2*>(Sh + (size_t)arow * Nn + ks + kofs);
    v2f_t A = { av.x, av.y };
    const float m0 = -kkh[ks + kofs];
    const float m1 = -kkh[ks + kofs + 1];
#pragma unroll
    for (int tt = 0; tt < 4; tt++) {
      // B fragment of ab[m][j] = (-kk[m]) * (kk[j]*a[j]) — built in-register
      v2f_t B = { m0 * cv[tt], m1 * cv[tt] };
      acc[tt] = __builtin_amdgcn_wmma_f32_16x16x4_f32(
          false, A, false, B, (short)0, acc[tt], false, false);
    }
  }

  __shared__ float outsm[Nn];
  __shared__ float r1[Nn], r2[Nn];
  __shared__ float gstat[2];

  float vrow[8];
#pragma unroll
  for (int p = 0; p < 8; p++) vrow[p] = vch[i0 + half * 8 + p];
  float part[8];
#pragma unroll
  for (int p = 0; p < 8; p++) part[p] = 0.f;
#pragma unroll
  for (int p = 0; p < 8; p++) {
    const int i = i0 + half * 8 + p;        // D tile row mapping
#pragma unroll
    for (int tt = 0; tt < 4; tt++) {
      const int j = tt * 16 + lr;
      const float sold = Sh[(size_t)i * Nn + j];
      const float sn = sold * wv[tt] + acc[tt][p] + vrow[p] * kv[tt];
      So[(size_t)i * Nn + j] = sn;
      part[p] += sn * rv[tt];
    }
  }
#pragma unroll
  for (int p = 0; p < 8; p++) {
#pragma unroll
    for (int m = 1; m < 16; m <<= 1) part[p] += __shfl_xor(part[p], m, 32);
  }
  if (lr == 0) {
#pragma unroll
    for (int p = 0; p < 8; p++) outsm[i0 + half * 8 + p] = part[p];
  }
  __syncthreads();
  if (t < Nn) { const float v = outsm[t]; r1[t] = v; r2[t] = v * v; }
  __syncthreads();
  for (int o = 32; o > 0; o >>= 1) {
    if (t < o) { r1[t] += r1[t + o]; r2[t] += r2[t + o]; }
    __syncthreads();
  }
  if (t == 0) {
    const float mu = r1[0] * (1.f / Nn);
    gstat[0] = mu;
    gstat[1] = rsqrtf(r2[0] * (1.f / Nn) - mu * mu + 0.00064f);
  }
  __syncthreads();
  if (t < Nn) {
    const int c = h * Nn + t;
    float o = (outsm[t] - gstat[0]) * gstat[1] * lnx_w[(size_t)l * Cc + c] + lnx_b[(size_t)l * Cc + c];
    o += ws[W_RKS + h] * vch[t];
    ws[W_OG + c] = o * ws[W_G + c];
  }
}

// ln2 + FFN token shift; writes new_x2 and kfin.
__global__ void k_ln2prep(int l, const float* __restrict__ ln2_w, const float* __restrict__ ln2_b,
                          const float* __restrict__ s_x2, const float* __restrict__ ffn_mx,
                          float* __restrict__ out_x2, float* __restrict__ ws) {
  __shared__ float red[256];
  const int t = threadIdx.x;
  float v[4], s = 0.f, ss = 0.f;
#pragma unroll
  for (int q = 0; q < 4; q++) { v[q] = ws[W_X + t * 4 + q]; s += v[q]; ss += v[q] * v[q]; }
  float mu, rs; ln_stats_256(s, ss, red, mu, rs, 1e-5f);
  const size_t lC = (size_t)l * Cc;
#pragma unroll
  for (int q = 0; q < 4; q++) {
    const int c = t * 4 + q;
    const float xn = (v[q] - mu) * rs * ln2_w[lC + c] + ln2_b[lC + c];
    out_x2[lC + c] = xn;
    ws[W_KFIN + c] = xn + (s_x2[lC + c] - xn) * ffn_mx[lC + c];
  }
}

__global__ void k_finalln(const float* __restrict__ w, const float* __restrict__ b,
                          float* __restrict__ ws) {
  __shared__ float red[256];
  const int t = threadIdx.x;
  float v[4], s = 0.f, ss = 0.f;
#pragma unroll
  for (int q = 0; q < 4; q++) { v[q] = ws[W_X + t * 4 + q]; s += v[q]; ss += v[q] * v[q]; }
  float mu, rs; ln_stats_256(s, ss, red, mu, rs, 1e-5f);
#pragma unroll
  for (int q = 0; q < 4; q++) {
    const int c = t * 4 + q;
    ws[W_XH + c] = (v[q] - mu) * rs * w[c] + b[c];
  }
}

extern "C" void kernel_launch(void* const* d_in, const int* in_sizes, int n_in,
                              void* d_out, int out_size, void* d_ws, size_t ws_size,
                              hipStream_t stream) {
  (void)in_sizes; (void)n_in; (void)out_size; (void)ws_size;
  const int*   p_token = (const int*)  d_in[0];
  const float* p_emb   = (const float*)d_in[1];
  const float* p_ln0w  = (const float*)d_in[2];
  const float* p_ln0b  = (const float*)d_in[3];
  const float* p_ln1w  = (const float*)d_in[4];
  const float* p_ln1b  = (const float*)d_in[5];
  const float* p_ln2w  = (const float*)d_in[6];
  const float* p_ln2b  = (const float*)d_in[7];
  const float* p_mxr   = (const float*)d_in[8];
  const float* p_mxw   = (const float*)d_in[9];
  const float* p_mxk   = (const float*)d_in[10];
  const float* p_mxv   = (const float*)d_in[11];
  const float* p_mxa   = (const float*)d_in[12];
  const float* p_mxg   = (const float*)d_in[13];
  const float* p_w0    = (const float*)d_in[14];
  const float* p_w1    = (const float*)d_in[15];
  const float* p_w2    = (const float*)d_in[16];
  const float* p_a0    = (const float*)d_in[17];
  const float* p_a1    = (const float*)d_in[18];
  const float* p_a2    = (const float*)d_in[19];
  const float* p_v0    = (const float*)d_in[20];
  const float* p_v1    = (const float*)d_in[21];
  const float* p_v2    = (const float*)d_in[22];
  const float* p_g1    = (const float*)d_in[23];
  const float* p_g2    = (const float*)d_in[24];
  const float* p_kk    = (const float*)d_in[25];
  const float* p_ka    = (const float*)d_in[26];
  const float* p_rk    = (const float*)d_in[27];
  const float* p_attkw = (const float*)d_in[28];
  const float* p_attvw = (const float*)d_in[29];
  const float* p_attrw = (const float*)d_in[30];
  const float* p_attow = (const float*)d_in[31];
  const float* p_lnxw  = (const float*)d_in[32];
  const float* p_lnxb  = (const float*)d_in[33];
  const float* p_ffnmx = (const float*)d_in[34];
  const float* p_ffnkw = (const float*)d_in[35];
  const float* p_ffnvw = (const float*)d_in[36];
  const float* p_lnow  = (const float*)d_in[37];
  const float* p_lnob  = (const float*)d_in[38];
  const float* p_headw = (const float*)d_in[39];
  const float* p_sx1   = (const float*)d_in[40];
  const float* p_sS    = (const float*)d_in[41];
  const float* p_sx2   = (const float*)d_in[42];

  float* out = (float*)d_out;
  float* ws  = (float*)d_ws;

  k_embed<<<1, 256, 0, stream>>>(p_token, p_emb, p_ln0w, p_ln0b, ws);

  for (int l = 0; l < Ll; l++) {
    const size_t lCC = (size_t)l * Cc * Cc;
    k_prep<<<1, 256, 0, stream>>>(l, p_ln1w, p_ln1b, p_sx1,
                                  p_mxr, p_mxw, p_mxk, p_mxv, p_mxa, p_mxg,
                                  out + O_X1, ws);
    k_lora1<<<4, 512, 0, stream>>>(l, p_w1, p_a1, p_v1, p_g1, ws);
    k_lora2<<<16, 256, 0, stream>>>(l, p_w2, p_a2, p_v2, p_g2, p_w0, p_a0, p_v0, ws);
    k_gemv3<<<384, 256, 0, stream>>>(p_attrw + lCC, p_attkw + lCC, p_attvw + lCC, ws);
    k_post1<<<1, 256, 0, stream>>>(l, p_kk, p_ka, p_rk, ws);
    k_state<<<Hh, 128, 0, stream>>>(l, p_sS, p_lnxw, p_lnxb, out + O_S, ws);
    // x += ow @ (out*g)
    k_gemv<<<Cc / 8, 256, 0, stream>>>(p_attow + lCC, ws + W_OG, ws + W_X, ws + W_X, Cc / 4, 1);
    k_ln2prep<<<1, 256, 0, stream>>>(l, p_ln2w, p_ln2b, p_sx2, p_ffnmx, out + O_X2, ws);
    // kf = relu(ffn_kw @ kfin)^2
    k_gemv<<<FFd / 8, 256, 0, stream>>>(p_ffnkw + (size_t)l * FFd * Cc, ws + W_KFIN,
                                        nullptr, ws + W_KF, Cc / 4, 2);
    // x += ffn_vw @ kf
    k_gemv<<<Cc / 8, 256, 0, stream>>>(p_ffnvw + (size_t)l * Cc * FFd, ws + W_KF,
                                       ws + W_X, ws + W_X, FFd / 4, 1);
  }

  k_finalln<<<1, 256, 0, stream>>>(p_lnow, p_lnob, ws);
  // logits = head_w @ xh   (65536 rows, the 256 MB bandwidth-dominant GEMV)
  k_gemv<<<Vv / 8, 256, 0, stream>>>(p_headw, ws + W_XH, nullptr, out + O_LOGITS, Cc / 4, 0);
}